// MinDeltaRSumHead_30253749633427
// MI455X (gfx1250) — compile-verified
//
#include <hip/hip_runtime.h>
#include <math.h>

// MI455X / gfx1250: wave32. Async global->LDS staging in, async LDS->global out.
// One thread per event (B=400000, 10 jets x float4).

#define THREADS      256
#define NJ           10
#define NPAIR        45
#define NCAND        630
#define EV_STRIDE_W  44            // 176 B per event slice in LDS (16B aligned)

// ---------- compile-time candidate table (exact SUMS lexicographic order) ----------
struct CandTable {
    short         pa[NCAND];       // first pair index into g[]
    short         pb[NCAND];       // second pair index into g[]
    int           packed[NCAND];   // (i0<<12)|(j0<<8)|(i1<<4)|j1 jet labels
};

constexpr CandTable makeCandTable() {
    CandTable t{};
    int P0[NPAIR] = {}, P1[NPAIR] = {};
    int p = 0;
    for (int i = 0; i < NJ; ++i)
        for (int j = i + 1; j < NJ; ++j) { P0[p] = i; P1[p] = j; ++p; }
    int n = 0;
    for (int a = 0; a < NPAIR; ++a)
        for (int b = a + 1; b < NPAIR; ++b)
            if (P0[a] != P0[b] && P0[a] != P1[b] && P1[a] != P0[b] && P1[a] != P1[b]) {
                t.pa[n] = (short)a;
                t.pb[n] = (short)b;
                t.packed[n] = (P0[a] << 12) | (P1[a] << 8) | (P0[b] << 4) | P1[b];
                ++n;
            }
    return t;
}
constexpr CandTable CT = makeCandTable();

// quarter boundaries (contiguous => first-occurrence tie-break preserved on merge)
#define Q0 0
#define Q1 158
#define Q2 316
#define Q3 474
#define Q4 NCAND

__device__ __forceinline__ float wrap_dphi(float dphi) {
    const float PI_F      = 3.14159265358979323846f;
    const float TWO_PI_F  = 6.28318530717958647692f;
    const float INV_2PI_F = 0.15915494309189533577f;
    float t = dphi + PI_F;                       // (dphi + pi) % (2pi) - pi, floor-mod
    t -= floorf(t * INV_2PI_F) * TWO_PI_F;
    return t - PI_F;
}

__global__ __launch_bounds__(THREADS)
void mindr_kernel(const float* __restrict__ x, float* __restrict__ out,
                  int B, unsigned totalChunks) {
    __shared__ __align__(16) float lds[THREADS * EV_STRIDE_W];

    const unsigned tid = threadIdx.x;
    const unsigned blockChunk0 = blockIdx.x * (unsigned)(THREADS * NJ);
    const float* base = x + (size_t)blockIdx.x * (THREADS * NJ * 4);   // uniform -> SGPR pair
    const unsigned lbase = (unsigned)(size_t)(&lds[0]);                // LDS byte offset

    // ================= async stage-in: 256 events * 160B, coalesced 16B chunks ==========
    if (blockChunk0 + (THREADS * NJ) <= totalChunks) {                  // full block (fast path)
#pragma unroll
        for (int k = 0; k < NJ; ++k) {
            unsigned c = (unsigned)k * THREADS + tid;
            unsigned evl = c / NJ, r = c - evl * NJ;
            unsigned ldsoff = lbase + evl * (EV_STRIDE_W * 4) + r * 16;
            unsigned goff   = c * 16;
            asm volatile("global_load_async_to_lds_b128 %0, %1, %2"
                         :: "v"(ldsoff), "v"(goff), "s"(base) : "memory");
        }
    } else {                                                            // tail block
#pragma unroll
        for (int k = 0; k < NJ; ++k) {
            unsigned c = (unsigned)k * THREADS + tid;
            if (blockChunk0 + c < totalChunks) {
                unsigned evl = c / NJ, r = c - evl * NJ;
                unsigned ldsoff = lbase + evl * (EV_STRIDE_W * 4) + r * 16;
                unsigned goff   = c * 16;
                asm volatile("global_load_async_to_lds_b128 %0, %1, %2"
                             :: "v"(ldsoff), "v"(goff), "s"(base) : "memory");
            }
        }
    }
    asm volatile("s_wait_asynccnt 0" ::: "memory");
    __syncthreads();

    const int ev = (int)(blockIdx.x * THREADS + tid);
    const int wb = (int)tid * EV_STRIDE_W;

    if (ev < B) {
        // ---- per-jet kinematics ----
        float eta[NJ], phi[NJ];
#pragma unroll
        for (int n = 0; n < NJ; ++n) {
            float4 v = *reinterpret_cast<const float4*>(&lds[wb + 4 * n]); // [E,px,py,pz]
            float pt = sqrtf(v.y * v.y + v.z * v.z);
            eta[n] = asinhf(v.w / pt);
            phi[n] = atan2f(v.z, v.y);
        }
        // stash eta/phi in own slice for dynamic-index fetch at the end
#pragma unroll
        for (int n = 0; n < NJ; ++n) {
            lds[wb + n]      = eta[n];
            lds[wb + NJ + n] = phi[n];
        }

        // ---- g[p] = |dR(pair p) - 0.8| (registers, static indices) ----
        float g[NPAIR];
        {
            int p = 0;
#pragma unroll
            for (int i = 0; i < NJ; ++i)
#pragma unroll
                for (int j = i + 1; j < NJ; ++j) {
                    float de = eta[i] - eta[j];
                    float dp = wrap_dphi(phi[i] - phi[j]);
                    float d  = sqrtf(de * de + dp * dp);
                    g[p++]   = fabsf(d - 0.8f);
                }
        }

        // ---- argmin over 630 candidates; 4 contiguous quarters for ILP ----
        // NaN m never wins (strict <) == reference NaN->inf; init = SUMS[0].
        float b0 = __builtin_inff(), b1 = __builtin_inff(),
              b2 = __builtin_inff(), b3 = __builtin_inff();
        int   x0 = CT.packed[0], x1 = 0, x2 = 0, x3 = 0;
#pragma unroll
        for (int c = Q0; c < Q1; ++c) {
            float m = g[CT.pa[c]] + g[CT.pb[c]];
            bool lt = m < b0; b0 = lt ? m : b0; x0 = lt ? CT.packed[c] : x0;
        }
#pragma unroll
        for (int c = Q1; c < Q2; ++c) {
            float m = g[CT.pa[c]] + g[CT.pb[c]];
            bool lt = m < b1; b1 = lt ? m : b1; x1 = lt ? CT.packed[c] : x1;
        }
#pragma unroll
        for (int c = Q2; c < Q3; ++c) {
            float m = g[CT.pa[c]] + g[CT.pb[c]];
            bool lt = m < b2; b2 = lt ? m : b2; x2 = lt ? CT.packed[c] : x2;
        }
#pragma unroll
        for (int c = Q3; c < Q4; ++c) {
            float m = g[CT.pa[c]] + g[CT.pb[c]];
            bool lt = m < b3; b3 = lt ? m : b3; x3 = lt ? CT.packed[c] : x3;
        }
        // ordered merge (strict <): earlier quarter wins ties -> argmin semantics
        float best = b0; int bidx = x0;
        if (b1 < best) { best = b1; bidx = x1; }
        if (b2 < best) { best = b2; bidx = x2; }
        if (b3 < best) { best = b3; bidx = x3; }

        // ---- decode winner, recompute its two dR via dynamic LDS reads ----
        int i0 = (bidx >> 12) & 15, j0 = (bidx >> 8) & 15;
        int i1 = (bidx >> 4) & 15,  j1 = bidx & 15;

        float de0 = lds[wb + i0] - lds[wb + j0];
        float dp0 = wrap_dphi(lds[wb + NJ + i0] - lds[wb + NJ + j0]);
        float dr0 = sqrtf(de0 * de0 + dp0 * dp0);

        float de1 = lds[wb + i1] - lds[wb + j1];
        float dp1 = wrap_dphi(lds[wb + NJ + i1] - lds[wb + NJ + j1]);
        float dr1 = sqrtf(de1 * de1 + dp1 * dp1);

        // park the 7 results in own LDS slice for the async stage-out
        lds[wb + 0] = (float)i0;
        lds[wb + 1] = (float)j0;
        lds[wb + 2] = (float)i1;
        lds[wb + 3] = (float)j1;
        lds[wb + 4] = dr0;
        lds[wb + 5] = dr1;
        lds[wb + 6] = best;
    }

    __syncthreads();   // all results in LDS before cross-thread stage-out

    // ================= async stage-out: 256 events * 7 floats, coalesced ===============
    {
        const float* obase_c = out + (size_t)blockIdx.x * (THREADS * 7);
        const unsigned oTotal  = (unsigned)B * 7u;
        const unsigned oBlock0 = blockIdx.x * (unsigned)(THREADS * 7);
        if (oBlock0 + (THREADS * 7) <= oTotal) {                        // full block
#pragma unroll
            for (int k = 0; k < 7; ++k) {
                unsigned q = (unsigned)k * THREADS + tid;
                unsigned el = q / 7u, r = q - el * 7u;
                unsigned ldsoff = lbase + el * (EV_STRIDE_W * 4) + r * 4;
                unsigned goff   = q * 4;
                asm volatile("global_store_async_from_lds_b32 %0, %1, %2"
                             :: "v"(goff), "v"(ldsoff), "s"(obase_c) : "memory");
            }
        } else {                                                        // tail block
#pragma unroll
            for (int k = 0; k < 7; ++k) {
                unsigned q = (unsigned)k * THREADS + tid;
                if (oBlock0 + q < oTotal) {
                    unsigned el = q / 7u, r = q - el * 7u;
                    unsigned ldsoff = lbase + el * (EV_STRIDE_W * 4) + r * 4;
                    unsigned goff   = q * 4;
                    asm volatile("global_store_async_from_lds_b32 %0, %1, %2"
                                 :: "v"(goff), "v"(ldsoff), "s"(obase_c) : "memory");
                }
            }
        }
        asm volatile("s_wait_asynccnt 0" ::: "memory");
    }
}

extern "C" void kernel_launch(void* const* d_in, const int* in_sizes, int n_in,
                              void* d_out, int out_size, void* d_ws, size_t ws_size,
                              hipStream_t stream) {
    const float* x = (const float*)d_in[0];
    float* out = (float*)d_out;
    const int B = in_sizes[0] / (NJ * 4);              // 400000
    const unsigned totalChunks = (unsigned)B * NJ;     // 16B jet chunks
    const int nBlocks = (B + THREADS - 1) / THREADS;
    hipLaunchKernelGGL(mindr_kernel, dim3(nBlocks), dim3(THREADS), 0, stream,
                       x, out, B, totalChunks);
}